// GGL_57836029608468
// MI455X (gfx1250) — compile-verified
//
#include <hip/hip_runtime.h>
#include <hip/hip_bf16.h>
#include <math.h>

// CDNA5 / gfx1250: wave32, f32 WMMA 16x16x4 (the only full-f32 matrix op).
typedef __attribute__((ext_vector_type(2))) float v2f;
typedef __attribute__((ext_vector_type(8))) float v8f;

#define N_NODES 8
#define IN_FEAT 200
#define OUT_FEAT 10
#define EPS 1e-8f

// One wave (32 threads) does the entire pipeline:
//   atrr = sigmoid(x @ W^T + b)   via 50x v_wmma_f32_16x16x4_f32 (K=200)
//   dots = atrr @ atrr^T          via  3x v_wmma_f32_16x16x4_f32 (K=12, zero-padded)
//   cosine -> row-max normalize -> rank-based top-k (k=8) -> emit.
__global__ void __launch_bounds__(32)
gat_edge_fused(const float* __restrict__ x,   // (8, 200)
               const float* __restrict__ W,   // (10, 200)
               const float* __restrict__ b,   // (10,)
               float* __restrict__ out)       // 256 floats: values|edge_index|A_norm
{
    const int lane = threadIdx.x;   // 0..31
    const int half = lane >> 4;     // 0/1: selects K sub-pair (K0,K1) vs (K2,K3)
    const int lid  = lane & 15;     // A row / B col / C col index

    // Clamped per-lane base pointers; garbage rows/cols land only in C
    // rows >= 8 / cols >= 10, which are never read.
    const int mrow = lid & 7;
    const int ncol = (lid < OUT_FEAT) ? lid : 9;
    const float* xr = x + mrow * IN_FEAT + half * 2;
    const float* wr = W + ncol * IN_FEAT + half * 2;

    // ---------------- Stage 1: C = x @ W^T  (16x16 padded, K=200) ----------
    v8f acc = {};
    #pragma unroll 5
    for (int k4 = 0; k4 < IN_FEAT; k4 += 4) {
        const v2f a  = *(const v2f*)(xr + k4);    // global_load_b64
        const v2f bt = *(const v2f*)(wr + k4);    // global_load_b64
        acc = __builtin_amdgcn_wmma_f32_16x16x4_f32(
            false, a, false, bt, (short)0, acc, false, false);
    }

    // C/D layout: VGPR r -> (M=r + half*8, N=lid). Only M<8, N<10 are real.
    __shared__ float atrr[N_NODES][16];   // cols 10..15 zeroed -> exact K=12 Gram
    {
        const float bn = b[ncol];
        #pragma unroll
        for (int r = 0; r < 8; ++r) {
            float v = acc[r] + bn;
            v = __builtin_amdgcn_rcpf(1.0f + __expf(-v));   // fast sigmoid
            if (half == 0)
                atrr[r][lid] = (lid < OUT_FEAT) ? v : 0.0f;
        }
    }
    __syncthreads();

    // ---------------- Stage 2: dots = atrr @ atrr^T (K=12, zero-padded) ----
    // A row m=lid and B col n=lid are the same lane-local LDS address.
    const float* ar = &atrr[lid & 7][half * 2];
    v8f acc2 = {};
    #pragma unroll
    for (int k4 = 0; k4 < 12; k4 += 4) {
        const v2f a = *(const v2f*)(ar + k4);               // ds_load_b64
        acc2 = __builtin_amdgcn_wmma_f32_16x16x4_f32(
            false, a, false, a, (short)0, acc2, false, false);
    }

    __shared__ float dots[N_NODES][N_NODES];
    #pragma unroll
    for (int r = 0; r < 8; ++r)
        if (half == 0 && lid < N_NODES) dots[r][lid] = acc2[r];
    __syncthreads();

    // ---------------- Stage 3: cosine matrix (raw), norms ------------------
    __shared__ float norms[N_NODES];
    __shared__ float Araw[N_NODES][N_NODES];
    __shared__ float rinv[N_NODES];
    if (lane < N_NODES)
        norms[lane] = __builtin_amdgcn_sqrtf(dots[lane][lane]);
    __syncthreads();

    #pragma unroll
    for (int t = 0; t < 2; ++t) {          // 64 entries over 32 lanes
        const int e = lane + t * 32;
        const int i = e >> 3, j = e & 7;
        const float denom = fmaxf(norms[i] * norms[j], EPS);
        Araw[i][j] = dots[i][j] * __builtin_amdgcn_rcpf(denom);
    }
    __syncthreads();

    if (lane < N_NODES) {                  // per-row max -> reciprocal
        float m = Araw[lane][0];
        #pragma unroll
        for (int j = 1; j < N_NODES; ++j) m = fmaxf(m, Araw[lane][j]);
        rinv[lane] = __builtin_amdgcn_rcpf(m);
    }
    __syncthreads();

    // ------------- Stage 4: rank-based top-k (k=8) + emit ------------------
    // rinv[i] > 0, so ordering of A_norm row == ordering of Araw row.
    // rank(j) = #{c: A[c] > A[j]} + #{c<j: A[c] == A[j]}  (lax.top_k ties).
    #pragma unroll
    for (int t = 0; t < 2; ++t) {
        const int e = lane + t * 32;
        const int i = e >> 3, j = e & 7;
        const float vj = Araw[i][j];
        int rank = 0;
        #pragma unroll
        for (int c = 0; c < N_NODES; ++c) {
            const float vc = Araw[i][c];
            rank += (vc > vj) || ((vc == vj) && (c < j));
        }
        const float vn = vj * rinv[i];
        // Output (256 f32): values | edge_index[0] | edge_index[1] | A_norm
        out[        i * N_NODES + rank] = vn;        // sorted values
        out[ 64  +  e                 ] = (float)i;  // repeat(arange(8), 8)
        out[128  +  i * N_NODES + rank] = (float)j;  // sorted column indices
        out[192  +  e                 ] = vn;        // A_norm (original order)
    }
}

extern "C" void kernel_launch(void* const* d_in, const int* in_sizes, int n_in,
                              void* d_out, int out_size, void* d_ws, size_t ws_size,
                              hipStream_t stream) {
    (void)in_sizes; (void)n_in; (void)out_size; (void)d_ws; (void)ws_size;
    const float* x = (const float*)d_in[0];
    const float* W = (const float*)d_in[1];
    const float* b = (const float*)d_in[2];
    float* out = (float*)d_out;
    // Entire problem fits in one wave32 on one WGP; launch-latency bound.
    gat_edge_fused<<<1, 32, 0, stream>>>(x, W, b, out);
}